// SparseDMC_90890097918574
// MI455X (gfx1250) — compile-verified
//
#include <hip/hip_runtime.h>

#define NBLK 256
#define EPSF 1e-8f
#define WSCALE 0.99f

// ---------------------------------------------------------------------------
// CDNA5 async global->LDS staging (tracked by ASYNCcnt).
// Builtin existence proven by round-1 diagnostic; first param is
// 'int __vector(4) __device__ *'  i.e.  v4i addrspace(1)*.
// ---------------------------------------------------------------------------
typedef int v4i __attribute__((ext_vector_type(4)));
typedef __attribute__((address_space(1))) v4i gv4i_t;
typedef __attribute__((address_space(3))) v4i lv4i_t;

__device__ __forceinline__ void async_copy16(const void* g, void* l) {
#if __has_builtin(__builtin_amdgcn_global_load_async_to_lds_b128)
  // Per-lane 16B copy, global -> LDS, no VGPR data. Generic LDS pointers keep
  // the LDS byte offset in the low 32 bits, so integer truncation is the
  // correct generic->as(3) conversion; global generic == as(1).
  __builtin_amdgcn_global_load_async_to_lds_b128(
      (gv4i_t*)(unsigned long long)g,
      (lv4i_t*)(unsigned int)(unsigned long long)l,
      /*offset=*/0, /*cpol=*/0);
#else
  *(int4*)l = *(const int4*)g;
#endif
}

__device__ __forceinline__ void wait_async0() {
#if __has_builtin(__builtin_amdgcn_s_wait_asynccnt)
  __builtin_amdgcn_s_wait_asynccnt(0);
#elif defined(__AMDGCN__)
  asm volatile("s_wait_asynccnt 0" ::: "memory");
#endif
}

// ---------------------------------------------------------------------------
// Pass 3 (placed first so the disasm snippet shows the async staging):
// per-cube dual-MC vertex / normal / L_dev.
// beta/alpha/cube_idx are staged to LDS with coalesced async 16B/lane copies.
// Full blocks use pure linear addressing (offsets fold into IOFFSET); only
// the single tail block takes the guarded scalar-load fallback.
// ---------------------------------------------------------------------------
__global__ void __launch_bounds__(NBLK)
k_dmc(const long long* __restrict__ cidx, const float* __restrict__ sdf,
      const float* __restrict__ world, const float* __restrict__ beta,
      const float* __restrict__ alpha, float* __restrict__ vout,
      float* __restrict__ ldev, float* __restrict__ pout,
      float* __restrict__ nout, int n) {
  __shared__ long long s_idx[NBLK * 8];   // 16 KB
  __shared__ float     s_beta[NBLK * 12]; // 12 KB
  __shared__ float     s_alpha[NBLK * 8]; //  8 KB

  const int t = threadIdx.x;
  const long long base = (long long)blockIdx.x * NBLK;
  const long long c = base + t;

  if (base + NBLK <= (long long)n) {
    // Full block: coalesced async 16B/lane copies, linear addresses.
    const char* gi = (const char*)cidx + base * 64;
    const char* gb = (const char*)beta + base * 48;
    const char* ga = (const char*)alpha + base * 32;
    const int b16 = t * 16;
#pragma unroll
    for (int i = 0; i < 4; ++i)  // 16 KB of cube_idx
      async_copy16(gi + b16 + i * (NBLK * 16),
                   (char*)s_idx + b16 + i * (NBLK * 16));
#pragma unroll
    for (int i = 0; i < 3; ++i)  // 12 KB of beta
      async_copy16(gb + b16 + i * (NBLK * 16),
                   (char*)s_beta + b16 + i * (NBLK * 16));
#pragma unroll
    for (int i = 0; i < 2; ++i)  // 8 KB of alpha
      async_copy16(ga + b16 + i * (NBLK * 16),
                   (char*)s_alpha + b16 + i * (NBLK * 16));
    wait_async0();
  } else if (c < n) {
    // Tail block (1 of ~1954): plain guarded loads into LDS.
#pragma unroll
    for (int k = 0; k < 8; ++k) s_idx[t * 8 + k] = cidx[c * 8 + k];
#pragma unroll
    for (int e = 0; e < 12; ++e) s_beta[t * 12 + e] = beta[c * 12 + e];
#pragma unroll
    for (int k = 0; k < 8; ++k) s_alpha[t * 8 + k] = alpha[c * 8 + k];
  }
  __syncthreads();

  if (c >= n) return;

  long long idx[8];
#pragma unroll
  for (int k = 0; k < 8; ++k) idx[k] = s_idx[t * 8 + k];

  float sv[8];
#pragma unroll
  for (int k = 0; k < 8; ++k) sv[k] = sdf[idx[k]];

  bool anyP = false, anyN = false;
#pragma unroll
  for (int k = 0; k < 8; ++k) {
    if (sv[k] > 0.f) anyP = true; else anyN = true;
  }
  const bool surf = anyP && anyN;
  const float smask = surf ? 1.f : 0.f;

  float wx[8], wy[8], wz[8];
#pragma unroll
  for (int k = 0; k < 8; ++k) {
    const float* p = world + idx[k] * 3ll;
    wx[k] = p[0]; wy[k] = p[1]; wz[k] = p[2];
  }

  float an[8];
#pragma unroll
  for (int k = 0; k < 8; ++k)
    an[k] = tanhf(s_alpha[t * 8 + k]) * WSCALE + 1.f;

  const int E0[12] = {0, 1, 4, 0, 2, 3, 6, 2, 2, 3, 7, 6};
  const int E1[12] = {1, 5, 5, 4, 3, 7, 7, 6, 0, 1, 5, 4};

  float uex[12], uey[12], uez[12], cmv[12];
  float wsum = 0.f, vdx = 0.f, vdy = 0.f, vdz = 0.f;
  float nax = 0.f, nay = 0.f, naz = 0.f;
#pragma unroll
  for (int e = 0; e < 12; ++e) {
    const int i0 = E0[e], i1 = E1[e];
    const float s0 = sv[i0], s1 = sv[i1];
    const bool cr = ((s0 > 0.f) != (s1 > 0.f)) && surf;
    const float cm = cr ? 1.f : 0.f;
    cmv[e] = cm;
    const float bn = tanhf(s_beta[t * 12 + e]) * WSCALE + 1.f;
    const float w = bn * cm;  // wbe
    const float w0 = an[i0] * fabsf(s0);
    const float w1 = an[i1] * fabsf(s1);
    const float inv = 1.f / (w0 + w1 + EPSF);
    const float p0x = wx[i0], p0y = wy[i0], p0z = wz[i0];
    const float p1x = wx[i1], p1y = wy[i1], p1z = wz[i1];
    const float ux = (p0x * w1 + p1x * w0) * inv;
    const float uy = (p0y * w1 + p1y * w0) * inv;
    const float uz = (p0z * w1 + p1z * w0) * inv;
    uex[e] = ux; uey[e] = uy; uez[e] = uz;
    wsum += w;
    vdx += ux * w; vdy += uy * w; vdz += uz * w;
    const float evx = p1x - p0x, evy = p1y - p0y, evz = p1z - p0z;
    const float elen = sqrtf(evx * evx + evy * evy + evz * evz) + EPSF;
    const float d = s1 - s0;
    const float sg = (d > 0.f) ? 1.f : ((d < 0.f) ? -1.f : 0.f);
    const float f = (sg / elen) * w;
    nax += evx * f; nay += evy * f; naz += evz * f;
  }

  const float invw = 1.f / (wsum + EPSF);
  vdx *= invw; vdy *= invw; vdz *= invw;
  nax *= invw; nay *= invw; naz *= invw;
  const float nl = sqrtf(nax * nax + nay * nay + naz * naz) + EPSF;
  nax /= nl; nay /= nl; naz /= nl;

  float dist[12];
  float dsum = 0.f, nedges = 0.f;
#pragma unroll
  for (int e = 0; e < 12; ++e) {
    const float dx = uex[e] - vdx, dy = uey[e] - vdy, dz = uez[e] - vdz;
    const float di = sqrtf(dx * dx + dy * dy + dz * dz) * cmv[e];
    dist[e] = di; dsum += di; nedges += cmv[e];
  }
  const float mean = dsum / (nedges + EPSF);
#pragma unroll
  for (int e = 0; e < 12; ++e)
    ldev[c * 12 + e] = fabsf(dist[e] - mean) * cmv[e];

  const float vx = vdx * smask, vy = vdy * smask, vz = vdz * smask;
  vout[c * 3 + 0] = vx; vout[c * 3 + 1] = vy; vout[c * 3 + 2] = vz;
  pout[c * 3 + 0] = vx; pout[c * 3 + 1] = vy; pout[c * 3 + 2] = vz;
  nout[c * 3 + 0] = nax * smask;
  nout[c * 3 + 1] = nay * smask;
  nout[c * 3 + 2] = naz * smask;
}

// ---------------------------------------------------------------------------
// Pass 1: world = -1 + deform  (corners never scattered keep unique_pos == 0)
// ---------------------------------------------------------------------------
__global__ void k_world_init(const float* __restrict__ deform,
                             float* __restrict__ world, int m3) {
  int i = blockIdx.x * blockDim.x + threadIdx.x;
  if (i < m3) world[i] = deform[i] - 1.0f;
}

// ---------------------------------------------------------------------------
// Pass 2: scatter corner positions: world[idx] = pos*(2/res) - 1 + deform[idx]
// One thread per (cube, corner). Duplicate indices race (reference scatter
// order is also unspecified).
// ---------------------------------------------------------------------------
__global__ void k_scatter(const int* __restrict__ vox,
                          const long long* __restrict__ cidx,
                          const float* __restrict__ deform,
                          const int* __restrict__ resP,
                          float* __restrict__ world, int n) {
  int j = blockIdx.x * blockDim.x + threadIdx.x;  // j = c*8 + k
  if (j >= n * 8) return;
  int c = j >> 3, k = j & 7;
  float scale = 2.0f / (float)resP[0];
  long long m = cidx[j];
  float px = (float)(vox[c * 3 + 0] + (k & 1));
  float py = (float)(vox[c * 3 + 1] + ((k >> 1) & 1));
  float pz = (float)(vox[c * 3 + 2] + ((k >> 2) & 1));
  world[m * 3 + 0] = px * scale - 1.0f + deform[m * 3 + 0];
  world[m * 3 + 1] = py * scale - 1.0f + deform[m * 3 + 1];
  world[m * 3 + 2] = pz * scale - 1.0f + deform[m * 3 + 2];
}

// ---------------------------------------------------------------------------
extern "C" void kernel_launch(void* const* d_in, const int* in_sizes, int n_in,
                              void* d_out, int out_size, void* d_ws,
                              size_t ws_size, hipStream_t stream) {
  const int*       vox    = (const int*)d_in[0];        // voxel_coords N*3 i32
  const float*     sdf    = (const float*)d_in[1];      // sdf M f32
  const long long* cidx   = (const long long*)d_in[2];  // cube_idx N*8 i64
  const int*       resP   = (const int*)d_in[3];        // resolution scalar
  const float*     deform = (const float*)d_in[4];      // deform M*3 f32
  const float*     beta   = (const float*)d_in[5];      // beta N*12 f32
  const float*     alpha  = (const float*)d_in[6];      // alpha N*8 f32
  // d_in[7] = gamma is mathematically dead in the reference (gamma_n*0+1 == 1)

  const int N = in_sizes[0] / 3;
  const int M = in_sizes[1];
  const int M3 = M * 3;

  float* world = (float*)d_ws;  // M*3 floats scratch (7.2 MB)

  float* out      = (float*)d_out;
  float* vertices = out;                       // N*3
  float* ldev     = out + (size_t)N * 3;       // N*12 (faces is empty)
  float* palpha   = out + (size_t)N * 15;      // N*3
  float* nrm      = out + (size_t)N * 18;      // N*3

  k_world_init<<<(M3 + 255) / 256, 256, 0, stream>>>(deform, world, M3);
  k_scatter<<<(N * 8 + 255) / 256, 256, 0, stream>>>(vox, cidx, deform, resP,
                                                     world, N);
  k_dmc<<<(N + NBLK - 1) / NBLK, NBLK, 0, stream>>>(cidx, sdf, world, beta,
                                                    alpha, vertices, ldev,
                                                    palpha, nrm, N);
}